// OneHotMLPLSV_TopK_19000935317809
// MI455X (gfx1250) — compile-verified
//
#include <hip/hip_runtime.h>
#include <hip/hip_bf16.h>

typedef __attribute__((ext_vector_type(2))) float v2f;
typedef __attribute__((ext_vector_type(8))) float v8f;

#define T_TOK   32768      // B*S
#define DIM     1024       // D
#define NEXP    16         // E
#define HID     64         // H
#define WROWS   32         // gathered rows (M) per wave
#define TILE_M  256        // tokens per expert block (8 waves * 32)
#define MAX_TILES (T_TOK / TILE_M)   // 128
#define CHUNK   64         // K-chunk of D staged in LDS
#define XPITCH  96         // dwords per X/H pair-row (32-bank shift per pair)
#define WPITCH  160        // dwords per W pair-row  (32-bank shift per pair)

// ---------------------------------------------------------------------------
// Kernel 1: router. One wave per token: logits = x[t]·Wr + br, top-2 pick,
// append token to both experts' lists.
// ---------------------------------------------------------------------------
__global__ __launch_bounds__(256) void router_kernel(
    const float* __restrict__ x, const float* __restrict__ Wr,
    const float* __restrict__ br, int* __restrict__ cnt,
    int* __restrict__ lists)
{
    const int wave = (blockIdx.x * blockDim.x + threadIdx.x) >> 5;
    const int lane = threadIdx.x & 31;
    if (wave >= T_TOK) return;
    const int t = wave;

    const float* xr = x + (size_t)t * DIM;
    float acc[NEXP];
#pragma unroll
    for (int e = 0; e < NEXP; e++) acc[e] = 0.f;

    for (int d = lane; d < DIM; d += 32) {
        const float xv = xr[d];
        const float* wr = Wr + (size_t)d * NEXP;
#pragma unroll
        for (int e = 0; e < NEXP; e++) acc[e] = fmaf(xv, wr[e], acc[e]);
    }
#pragma unroll
    for (int e = 0; e < NEXP; e++) {
        float v = acc[e];
#pragma unroll
        for (int off = 16; off > 0; off >>= 1) v += __shfl_xor(v, off, 32);
        acc[e] = v + br[e];
    }
    if (lane == 0) {
        int i0 = 0; float v0 = acc[0];
#pragma unroll
        for (int e = 1; e < NEXP; e++) if (acc[e] > v0) { v0 = acc[e]; i0 = e; }
        int i1 = -1; float v1 = -3.4e38f;
#pragma unroll
        for (int e = 0; e < NEXP; e++)
            if (e != i0 && acc[e] > v1) { v1 = acc[e]; i1 = e; }
        int p0 = atomicAdd(&cnt[i0], 1); lists[i0 * T_TOK + p0] = t;
        int p1 = atomicAdd(&cnt[i1], 1); lists[i1 * T_TOK + p1] = t;
    }
}

// ---------------------------------------------------------------------------
// Kernel 2: out = x (residual base for the scatter-adds)
// ---------------------------------------------------------------------------
__global__ __launch_bounds__(256) void residual_copy_kernel(
    const float4* __restrict__ x, float4* __restrict__ out, int n4)
{
    int i = blockIdx.x * blockDim.x + threadIdx.x;
    if (i < n4) out[i] = x[i];
}

// ---------------------------------------------------------------------------
// Kernel 3: per-expert fused MLP on gathered tokens, f32 WMMA 16x16x4.
// Block = 8 waves; wave w owns gathered rows [w*32, w*32+32).
// LDS layouts are K-pair interleaved: the two K rows of every WMMA fragment
// are adjacent (one aligned ds_load_b64 per operand), and pair-row pitches
// (96 / 160 dwords) shift consecutive pairs by 32 banks so lane-halves
// (pair p vs p+1) hit disjoint bank halves -> conflict-free wave-wide b64.
// ---------------------------------------------------------------------------
__global__ __launch_bounds__(256) void moe_expert_kernel(
    const float* __restrict__ x,  const float* __restrict__ W1,
    const float* __restrict__ b1, const float* __restrict__ W2,
    const float* __restrict__ b2, const int* __restrict__ cnt,
    const int* __restrict__ lists, float* __restrict__ out)
{
    __shared__ float Xp[8][CHUNK / 2][XPITCH]; // per-wave X chunk / H tile (96 KB)
    __shared__ float Wp[CHUNK / 2][WPITCH];    // shared W1/W2 chunk (20 KB)
    __shared__ int   toks[TILE_M];

    const int e    = blockIdx.x / MAX_TILES;
    const int tile = blockIdx.x % MAX_TILES;
    const int n_e  = cnt[e];
    const int base = tile * TILE_M;
    if (base >= n_e) return;                 // uniform: WMMA below runs EXEC=all-1s

    const int tid   = threadIdx.x;
    const int wid   = tid >> 5;
    const int lane  = tid & 31;
    const int l16   = lane & 15;
    const int khalf = (lane >> 4) * 2;       // K offset for this lane half
    const int mhalf = (lane >> 4) * 8;       // M offset for C/D rows

    {
        int gi = base + tid;
        toks[tid] = (gi < n_e) ? lists[e * T_TOK + gi] : lists[e * T_TOK + base];
    }
    __syncthreads();

    // ---------------- layer 1: H = relu(Xg @ W1[e] + b1[e]) ----------------
    v8f acc[2][4];
#pragma unroll
    for (int mi = 0; mi < 2; mi++)
#pragma unroll
        for (int nt = 0; nt < 4; nt++)
#pragma unroll
            for (int i = 0; i < 8; i++) acc[mi][nt][i] = 0.f;

    for (int k0 = 0; k0 < DIM; k0 += CHUNK) {
        __syncthreads();
        // stage this wave's 32 gathered rows, pair-interleaved over K
#pragma unroll 4
        for (int i = 0; i < 16; i++) {
            int row = i * 2 + (lane >> 4);
            const float4* xr =
                (const float4*)(x + (size_t)toks[wid * WROWS + row] * DIM + k0);
            float4 v = xr[l16];                         // cols 4*l16 .. 4*l16+3
            const float* vv = (const float*)&v;
#pragma unroll
            for (int j = 0; j < 4; j++) {
                int c = l16 * 4 + j;
                Xp[wid][c >> 1][row * 2 + (c & 1)] = vv[j];
            }
        }
        // stage W1[e] chunk [64 x 64], pair-interleaved over K rows
        const float* w1g = W1 + (size_t)e * DIM * HID + (size_t)k0 * HID;
#pragma unroll
        for (int i = 0; i < 4; i++) {
            int flat4 = tid + i * 256;                  // 0..1023
            int row = flat4 >> 4, c4 = flat4 & 15;
            float4 v = *(const float4*)(w1g + (size_t)row * HID + c4 * 4);
            const float* vv = (const float*)&v;
#pragma unroll
            for (int j = 0; j < 4; j++)
                Wp[row >> 1][(c4 * 4 + j) * 2 + (row & 1)] = vv[j];
        }
        __syncthreads();

#pragma unroll
        for (int kk = 0; kk < CHUNK; kk += 4) {
            const int p = (kk + khalf) >> 1;
            v2f a0 = *(const v2f*)&Xp[wid][p][l16 * 2];
            v2f a1 = *(const v2f*)&Xp[wid][p][(16 + l16) * 2];
#pragma unroll
            for (int nt = 0; nt < 4; nt++) {
                v2f b = *(const v2f*)&Wp[p][(nt * 16 + l16) * 2];
                acc[0][nt] = __builtin_amdgcn_wmma_f32_16x16x4_f32(
                    false, a0, false, b, (short)0, acc[0][nt], false, false);
                acc[1][nt] = __builtin_amdgcn_wmma_f32_16x16x4_f32(
                    false, a1, false, b, (short)0, acc[1][nt], false, false);
            }
        }
    }

    // bias + relu -> per-wave H tile (32 x 64), pair-interleaved over H,
    // aliasing this wave's Xp region (same-wave DS ops are in order).
    float* Hw = &Xp[wid][0][0];
#pragma unroll
    for (int nt = 0; nt < 4; nt++) {
        const int c = nt * 16 + l16;
        float b1v = b1[e * HID + c];
#pragma unroll
        for (int mi = 0; mi < 2; mi++)
#pragma unroll
            for (int r = 0; r < 8; r++) {
                int m = mi * 16 + r + mhalf;
                float v = acc[mi][nt][r] + b1v;
                Hw[(c >> 1) * XPITCH + m * 2 + (c & 1)] = v > 0.f ? v : 0.f;
            }
    }

    // ---------------- layer 2: Y = H @ W2[e] + b2[e], scatter-add ----------
    const float* W2e = W2 + (size_t)e * HID * DIM;
    for (int n0 = 0; n0 < DIM; n0 += 64) {
        __syncthreads();
        // stage W2[e][:, n0:n0+64] (64 x 64), pair-interleaved over H rows
#pragma unroll
        for (int i = 0; i < 4; i++) {
            int flat4 = tid + i * 256;
            int row = flat4 >> 4, c4 = flat4 & 15;
            float4 v = *(const float4*)(W2e + (size_t)row * DIM + n0 + c4 * 4);
            const float* vv = (const float*)&v;
#pragma unroll
            for (int j = 0; j < 4; j++)
                Wp[row >> 1][(c4 * 4 + j) * 2 + (row & 1)] = vv[j];
        }
        __syncthreads();

        v8f acc2[2][4];
#pragma unroll
        for (int mi = 0; mi < 2; mi++)
#pragma unroll
            for (int nt = 0; nt < 4; nt++)
#pragma unroll
                for (int i = 0; i < 8; i++) acc2[mi][nt][i] = 0.f;

#pragma unroll
        for (int kk = 0; kk < HID; kk += 4) {
            const int p = (kk + khalf) >> 1;
            v2f a0 = *(const v2f*)&Hw[p * XPITCH + l16 * 2];
            v2f a1 = *(const v2f*)&Hw[p * XPITCH + (16 + l16) * 2];
#pragma unroll
            for (int nt = 0; nt < 4; nt++) {
                v2f b = *(const v2f*)&Wp[p][(nt * 16 + l16) * 2];
                acc2[0][nt] = __builtin_amdgcn_wmma_f32_16x16x4_f32(
                    false, a0, false, b, (short)0, acc2[0][nt], false, false);
                acc2[1][nt] = __builtin_amdgcn_wmma_f32_16x16x4_f32(
                    false, a1, false, b, (short)0, acc2[1][nt], false, false);
            }
        }
        // scatter: out[tok] += y + b2; two experts can hit one token -> atomics
#pragma unroll
        for (int nt = 0; nt < 4; nt++) {
            int col = n0 + nt * 16 + l16;
            float b2v = b2[e * DIM + col];
#pragma unroll
            for (int mi = 0; mi < 2; mi++)
#pragma unroll
                for (int r = 0; r < 8; r++) {
                    int m  = mi * 16 + r + mhalf;
                    int gi = base + wid * WROWS + m;
                    if (gi < n_e) {
                        int tok = toks[wid * WROWS + m];
                        unsafeAtomicAdd(out + (size_t)tok * DIM + col,
                                        acc2[mi][nt][r] + b2v);
                    }
                }
        }
    }
}

// ---------------------------------------------------------------------------
extern "C" void kernel_launch(void* const* d_in, const int* in_sizes, int n_in,
                              void* d_out, int out_size, void* d_ws, size_t ws_size,
                              hipStream_t stream)
{
    const float* x  = (const float*)d_in[0];
    const float* W1 = (const float*)d_in[1];
    const float* b1 = (const float*)d_in[2];
    const float* W2 = (const float*)d_in[3];
    const float* b2 = (const float*)d_in[4];
    const float* Wr = (const float*)d_in[5];
    const float* br = (const float*)d_in[6];
    float* out = (float*)d_out;

    int* cnt   = (int*)d_ws;          // [E]
    int* lists = (int*)d_ws + 16;     // [E][T]

    hipMemsetAsync(cnt, 0, NEXP * sizeof(int), stream);

    router_kernel<<<T_TOK / 8, 256, 0, stream>>>(x, Wr, br, cnt, lists);

    const int n4 = (T_TOK * DIM) / 4;
    residual_copy_kernel<<<n4 / 256, 256, 0, stream>>>(
        (const float4*)x, (float4*)out, n4);

    moe_expert_kernel<<<NEXP * MAX_TILES, 256, 0, stream>>>(
        x, W1, b1, W2, b2, cnt, lists, out);
}